// CapsuleLayer_10857677324542
// MI455X (gfx1250) — compile-verified
//
#include <hip/hip_runtime.h>
#include <hip/hip_bf16.h>

typedef __attribute__((ext_vector_type(2))) float v2f;
typedef __attribute__((ext_vector_type(8))) float v8f;

#define B_   64
#define N_   2048
#define D_   16
#define M_   32
#define C_   16
#define ROUTINGS 3
#define EPS_ 1e-7f

#define THREADS 1024
#define WAVES (THREADS/32)
#define TILES_PER_WAVE (N_/16/WAVES)   // 2048 rows / 16 per tile / 32 waves = 4

// x rows padded to 20 floats (80B): 20*lane mod 64 covers 16 distinct banks ->
// conflict-free WMMA A-operand loads; 80B is 16B-aligned so async B128 chunks
// (4 per 64B row) land on aligned LDS addresses.
#define XS_STRIDE 20

// LDS layout (floats):
//   xs  [N_][XS_STRIDE]  40960  (160 KB)
//   Ws  [D_][M_][C_]      8192  ( 32 KB)
//   Ts  [D_][M_]           512  agreement matrix T (b = x @ T)
//   yun [D_][M_]           512  unnormalized y^T, cross-wave reduced
//   Zs  [M_]                32  softmax denominators
//   sv  [M_][C_]           512  s
//   vv  [M_][C_]           512  v = squash(s)
//   scl [M_]                32  squash scale
#define LDS_FLOATS (N_*XS_STRIDE + D_*M_*C_ + 512 + 512 + 32 + 512 + 512 + 32)

// xor-16 lane exchange via DS_SWIZZLE group-of-32 mode:
// and_mask=0x1f, or_mask=0, xor_mask=0x10  ->  offset = 0x401f  (SWAPX16)
__device__ __forceinline__ float swap16(float v) {
    return __int_as_float(__builtin_amdgcn_ds_swizzle(__float_as_int(v), 0x401f));
}

__global__ void __launch_bounds__(THREADS, 1)
caps_routing_kernel(const float* __restrict__ x,
                    const float* __restrict__ W,
                    float* __restrict__ out) {
    extern __shared__ float lds[];
    float* xs  = lds;                       // 40960
    float* Ws  = xs  + N_*XS_STRIDE;        // 8192
    float* Ts  = Ws  + D_*M_*C_;            // 512
    float* yun = Ts  + 512;                 // 512
    float* Zs  = yun + 512;                 // 32
    float* sv  = Zs  + 32;                  // 512
    float* vv  = sv  + 512;                 // 512
    float* scl = vv  + 512;                 // 32

    const int b    = blockIdx.x;
    const int tid  = threadIdx.x;
    const int lane = tid & 31;
    const int wave = tid >> 5;
    const int lm   = lane & 15;   // lane within half
    const int lg   = lane >> 4;   // lane group (0: lanes 0-15, 1: lanes 16-31)

    // ---- async bulk fill: x[b] (128KB) and W (32KB), HBM/L2 -> LDS directly ----
    {
        const uint32_t xs_lds = (uint32_t)(uintptr_t)xs;   // low 32b of flat addr = LDS byte offset
        const uint64_t xgb    = (uint64_t)(uintptr_t)(x + (size_t)b * (N_*D_));
        #pragma unroll
        for (int k = 0; k < 8; ++k) {                      // 8192 16B chunks / 1024 threads
            const int c = tid + k*THREADS;                 // chunk id; row = c>>2, quarter = c&3
            const uint32_t voff = (uint32_t)c * 16u;       // global byte offset (rows contiguous)
            const uint32_t dsta = xs_lds + (uint32_t)(c >> 2) * (XS_STRIDE*4u)
                                         + (uint32_t)(c & 3) * 16u;
            asm volatile("global_load_async_to_lds_b128 %0, %1, %2"
                         :: "v"(dsta), "v"(voff), "s"(xgb) : "memory");
        }
        const uint32_t ws_lds = (uint32_t)(uintptr_t)Ws;
        const uint64_t wgb    = (uint64_t)(uintptr_t)W;
        #pragma unroll
        for (int k = 0; k < 2; ++k) {                      // 2048 16B chunks
            const int c = tid + k*THREADS;
            const uint32_t voff = (uint32_t)c * 16u;
            asm volatile("global_load_async_to_lds_b128 %0, %1, %2"
                         :: "v"(ws_lds + voff), "v"(voff), "s"(wgb) : "memory");
        }
        for (int i = tid; i < 512; i += THREADS) { Ts[i] = 0.f; yun[i] = 0.f; }
        if (tid < M_) Zs[tid] = 0.f;
        asm volatile("s_wait_asynccnt 0x0" ::: "memory");
    }
    __syncthreads();

    for (int it = 0; it < ROUTINGS; ++it) {
        // persistent per-wave partial of y_un^T[d][m]  (M-dim=d, N-dim=m halves)
        v8f yT0 = {}; v8f yT1 = {};
        float zp0 = 0.f, zp1 = 0.f;

        for (int t = 0; t < TILES_PER_WAVE; ++t) {
            const int n0 = (wave * TILES_PER_WAVE + t) * 16;

            // ---- GEMM1: btile(16n x 32m) = x_tile(16x16) @ T(16x32) ----
            // A layout (f32 16x4): lanes0-15 -> K=k0,k0+1 ; lanes16-31 -> K=k0+2,k0+3
            v8f e0 = {}; v8f e1 = {};
            #pragma unroll
            for (int k0 = 0; k0 < 16; k0 += 4) {
                const int kb = k0 + 2*lg;
                v2f a;  a.x  = xs[(n0+lm)*XS_STRIDE + kb];
                        a.y  = xs[(n0+lm)*XS_STRIDE + kb + 1];
                v2f b0; b0.x = Ts[kb*M_ + lm];          b0.y = Ts[(kb+1)*M_ + lm];
                v2f b1; b1.x = Ts[kb*M_ + 16 + lm];     b1.y = Ts[(kb+1)*M_ + 16 + lm];
                e0 = __builtin_amdgcn_wmma_f32_16x16x4_f32(false, a, false, b0, (short)0, e0, false, false);
                e1 = __builtin_amdgcn_wmma_f32_16x16x4_f32(false, a, false, b1, (short)0, e1, false, false);
            }

            // ---- e = exp(b) in-register (softmax numerator); Z partials ----
            #pragma unroll
            for (int r = 0; r < 8; ++r) {
                e0[r] = __expf(e0[r]); zp0 += e0[r];
                e1[r] = __expf(e1[r]); zp1 += e1[r];
            }

            // ---- GEMM2: yT(16d x 32m) += x^T_tile(16x16) @ e_tile(16x32) ----
            // B operand rebuilt from e's C-layout via xor-16 swizzles:
            //   C-layout: VGPR r = n_local r (lanes<16) / r+8 (lanes>=16)
            //   B-layout: lanes<16 -> n=k0,k0+1 ; lanes>=16 -> n=k0+2,k0+3
            #pragma unroll
            for (int k0 = 0; k0 < 16; k0 += 4) {
                const int kb = k0 + 2*lg;
                v2f a; a.x = xs[(n0+kb)*XS_STRIDE + lm];
                       a.y = xs[(n0+kb+1)*XS_STRIDE + lm];
                v2f b0, b1;
                if (k0 < 8) {
                    float p0 = swap16(e0[k0+2]);
                    float p1 = swap16(e0[k0+3]);
                    b0.x = lg ? p0 : e0[k0];
                    b0.y = lg ? p1 : e0[k0+1];
                    float q0 = swap16(e1[k0+2]);
                    float q1 = swap16(e1[k0+3]);
                    b1.x = lg ? q0 : e1[k0];
                    b1.y = lg ? q1 : e1[k0+1];
                } else {
                    const int r = k0 - 8;
                    float p0 = swap16(e0[r]);
                    float p1 = swap16(e0[r+1]);
                    b0.x = lg ? e0[r+2] : p0;
                    b0.y = lg ? e0[r+3] : p1;
                    float q0 = swap16(e1[r]);
                    float q1 = swap16(e1[r+1]);
                    b1.x = lg ? e1[r+2] : q0;
                    b1.y = lg ? e1[r+3] : q1;
                }
                yT0 = __builtin_amdgcn_wmma_f32_16x16x4_f32(false, a, false, b0, (short)0, yT0, false, false);
                yT1 = __builtin_amdgcn_wmma_f32_16x16x4_f32(false, a, false, b1, (short)0, yT1, false, false);
            }
        }

        // ---- cross-wave reduction via native LDS float atomics (ds_add_f32) ----
        atomicAdd(&Zs[lm],      zp0);
        atomicAdd(&Zs[16 + lm], zp1);
        #pragma unroll
        for (int r = 0; r < 8; ++r) {
            atomicAdd(&yun[(r + 8*lg)*M_ + lm],      yT0[r]);
            atomicAdd(&yun[(r + 8*lg)*M_ + 16 + lm], yT1[r]);
        }
        __syncthreads();

        // ---- s[m][c] = (1/Z[m]) * sum_d yun[d][m] * W[d][m][c] ----
        if (tid < M_*C_) {
            const int m = tid >> 4, cc = tid & 15;
            float acc = 0.f;
            #pragma unroll
            for (int d = 0; d < D_; ++d)
                acc += yun[d*M_ + m] * Ws[d*(M_*C_) + m*C_ + cc];
            sv[tid] = acc / Zs[m];
        }
        __syncthreads();

        // ---- squash scale per m ----
        if (tid < M_) {
            float n2 = 0.f;
            #pragma unroll
            for (int cc = 0; cc < C_; ++cc) { float v = sv[tid*C_ + cc]; n2 += v*v; }
            float n = sqrtf(n2);
            scl[tid] = (n2 / (1.f + n2)) / (n + EPS_);
        }
        __syncthreads();
        if (tid < M_*C_) vv[tid] = sv[tid] * scl[tid >> 4];
        __syncthreads();

        // ---- agreement fold: T[d][m] += sum_c v[m][c]*W[d][m][c]  (skip last) ----
        if (it + 1 < ROUTINGS) {
            if (tid < D_*M_) {
                const int d = tid >> 5, m = tid & 31;
                float acc = 0.f;
                #pragma unroll
                for (int cc = 0; cc < C_; ++cc)
                    acc += vv[m*C_ + cc] * Ws[d*(M_*C_) + m*C_ + cc];
                Ts[tid] += acc;
                yun[tid] = 0.f;     // reset reduction buffers for next iter
            }
            if (tid < M_) Zs[tid] = 0.f;
            __syncthreads();
        }
    }

    // ---- write v: out[b, m, c] ----
    if (tid < M_*C_) out[(size_t)b * (M_*C_) + tid] = vv[tid];
}

extern "C" void kernel_launch(void* const* d_in, const int* in_sizes, int n_in,
                              void* d_out, int out_size, void* d_ws, size_t ws_size,
                              hipStream_t stream) {
    const float* x  = (const float*)d_in[0];   // (64, 2048, 16) f32
    const float* W  = (const float*)d_in[1];   // (16, 32, 16)   f32
    float*      out = (float*)d_out;           // (64, 32, 16)   f32
    (void)in_sizes; (void)n_in; (void)out_size; (void)d_ws; (void)ws_size;
    const size_t smem = (size_t)LDS_FLOATS * sizeof(float);  // ~200 KB of 320 KB/WGP
    caps_routing_kernel<<<B_, THREADS, smem, stream>>>(x, W, out);
}